// MultiGat_57715770524245
// MI455X (gfx1250) — compile-verified
//
#include <hip/hip_runtime.h>
#include <math.h>

// ---------------------------------------------------------------------------
// MultiGAT forward for MI455X (gfx1250, wave32).
// GEMMs use V_WMMA_F32_16X16X4_F32 (fp32 matrix pipe, full precision),
// 4 independent accumulator tiles per wave (16x64 output strip) to cut A
// re-fetch 4x and keep the matrix pipe free of RAW chains.
// Graph aggregation is node-parallel over a device-built CSR (no fp atomics).
// ---------------------------------------------------------------------------

typedef __attribute__((ext_vector_type(2))) float v2f;
typedef __attribute__((ext_vector_type(8))) float v8f;

#define HC 256          // H*C
#define HEADS 4
#define CH 64           // channels per head
#define NEG_SLOPE 0.2f

// ---------------- small utility kernels ----------------

__global__ void k_zero_i32(int* p, int n) {
  int i = blockIdx.x * blockDim.x + threadIdx.x;
  if (i < n) p[i] = 0;
}

__global__ void k_copy_i32(const int* __restrict__ a, int* __restrict__ b, int n) {
  int i = blockIdx.x * blockDim.x + threadIdx.x;
  if (i < n) b[i] = a[i];
}

// X0[:,0:2] = pos, X0[:,2:64] = x   (reference: concat([pos, x], axis=1))
__global__ void k_concat(const float* __restrict__ x, const float* __restrict__ pos,
                         float* __restrict__ X0, int n) {
  int i = blockIdx.x * blockDim.x + threadIdx.x;
  if (i >= n * 64) return;
  int node = i >> 6, c = i & 63;
  X0[i] = (c < 2) ? pos[node * 2 + c] : x[node * 62 + (c - 2)];
}

// ---------------- CSR build over destination nodes ----------------

__global__ void k_count(const long long* __restrict__ dsts, int* __restrict__ cnt, int E) {
  int e = blockIdx.x * blockDim.x + threadIdx.x;
  if (e < E) atomicAdd(&cnt[(int)dsts[e]], 1);
}

// single-block exclusive scan of cnt[0..n) -> rowptr[0..n], rowptr[n] = total
__global__ void k_scan(const int* __restrict__ cnt, int* __restrict__ rowptr, int n) {
  __shared__ int part[256];
  int tid = threadIdx.x;
  int chunk = (n + 255) / 256;
  int lo = tid * chunk;
  int hi = lo + chunk; if (hi > n) hi = n; if (lo > n) lo = n;
  int s = 0;
  for (int i = lo; i < hi; ++i) s += cnt[i];
  part[tid] = s;
  __syncthreads();
  // Hillis-Steele inclusive scan over 256 partials
  for (int off = 1; off < 256; off <<= 1) {
    int v = part[tid];
    if (tid >= off) v += part[tid - off];
    __syncthreads();
    part[tid] = v;
    __syncthreads();
  }
  int run = (tid == 0) ? 0 : part[tid - 1];
  for (int i = lo; i < hi; ++i) { rowptr[i] = run; run += cnt[i]; }
  if (tid == 255) rowptr[n] = run;   // == E
}

__global__ void k_fill(const long long* __restrict__ srcs, const long long* __restrict__ dsts,
                       int* __restrict__ cursor, int* __restrict__ csr_src, int E) {
  int e = blockIdx.x * blockDim.x + threadIdx.x;
  if (e >= E) return;
  int d = (int)dsts[e];
  int p = atomicAdd(&cursor[d], 1);
  csr_src[p] = (int)srcs[e];
}

// ---------------- fp32 WMMA GEMM:  C[n_rows, 256] = A[n_rows, K] * W[K, 256] ----------------
// Block = 4 waves = one 16-row x 256-col output strip.
// Each wave holds FOUR v8f accumulators (16x64 strip): A fragment loaded once
// per k-step and reused across 4 WMMAs -> 4 independent dep chains, 4x less A
// traffic. Fragment layout per ISA: A 16x4 32-bit -> VGPR0 = {K0 | K2},
// VGPR1 = {K1 | K3} across lane halves; C/D: VGPR r = rows {r | r+8}.
__global__ void k_gemm_wmma(const float* __restrict__ A, const float* __restrict__ W,
                            float* __restrict__ C, int n_rows, int K) {
  const int wid  = threadIdx.x >> 5;           // 0..3 -> 64-col strip
  const int lane = threadIdx.x & 31;
  const int m0 = blockIdx.x * 16;
  if (m0 >= n_rows) return;                    // block-uniform: EXEC stays all-1s
  const int n0 = wid * 64;
  const int half = lane >> 4;                  // 0: lanes 0-15, 1: lanes 16-31
  const int l16  = lane & 15;

  v8f c0 = {}, c1 = {}, c2 = {}, c3 = {};
  const float* Arow = A + (size_t)(m0 + l16) * K;  // this lane's A row
  const float* Wb   = W + l16 + n0;                // this lane's base B column

  for (int k0 = 0; k0 < K; k0 += 4) {
    if (k0 + 16 < K) __builtin_prefetch(&Arow[k0 + 16], 0, 3);  // global_prefetch_b8
    v2f a;
    a.x = Arow[k0 + 2 * half + 0];
    a.y = Arow[k0 + 2 * half + 1];
    const float* w0 = Wb + (size_t)(k0 + 2 * half) * HC;   // row k, this col
    v2f b0, b1, b2, b3;
    b0.x = w0[0];  b0.y = w0[HC + 0];
    b1.x = w0[16]; b1.y = w0[HC + 16];
    b2.x = w0[32]; b2.y = w0[HC + 32];
    b3.x = w0[48]; b3.y = w0[HC + 48];
    c0 = __builtin_amdgcn_wmma_f32_16x16x4_f32(false, a, false, b0, (short)0, c0, false, false);
    c1 = __builtin_amdgcn_wmma_f32_16x16x4_f32(false, a, false, b1, (short)0, c1, false, false);
    c2 = __builtin_amdgcn_wmma_f32_16x16x4_f32(false, a, false, b2, (short)0, c2, false, false);
    c3 = __builtin_amdgcn_wmma_f32_16x16x4_f32(false, a, false, b3, (short)0, c3, false, false);
  }

  float* Crow = C + n0 + l16;
#pragma unroll
  for (int r = 0; r < 8; ++r) {
    size_t m = (size_t)(m0 + r + half * 8) * HC;
    Crow[m + 0]  = c0[r];
    Crow[m + 16] = c1[r];
    Crow[m + 32] = c2[r];
    Crow[m + 48] = c3[r];
  }
}

// ---------------- attention logit dot products ----------------
// as[n,h] = sum_c h[n,h,c]*a_src[h,c];  ad likewise
__global__ void k_alpha(const float* __restrict__ h, const float* __restrict__ a_src,
                        const float* __restrict__ a_dst, float* __restrict__ as,
                        float* __restrict__ ad, int n) {
  int i = blockIdx.x * blockDim.x + threadIdx.x;
  if (i >= n * HEADS) return;
  int node = i >> 2, hd = i & 3;
  const float* hr = h + (size_t)node * HC + hd * CH;
  const float* s1 = a_src + hd * CH;
  const float* s2 = a_dst + hd * CH;
  float u = 0.f, v = 0.f;
#pragma unroll 8
  for (int c = 0; c < CH; ++c) { float t = hr[c]; u += t * s1[c]; v += t * s2[c]; }
  as[i] = u;
  ad[i] = v;
}

// ---------------- wave-per-node softmax + aggregate ----------------
// lane owns 8 contiguous channels; head = lane>>3. Self loop handled inline.
__global__ void k_aggregate(const float* __restrict__ h, const float* __restrict__ as,
                            const float* __restrict__ ad, const int* __restrict__ rowptr,
                            const int* __restrict__ csr_src, const float* __restrict__ bias,
                            float* __restrict__ out, int n) {
  int wid  = threadIdx.x >> 5;
  int lane = threadIdx.x & 31;
  int node = blockIdx.x * (blockDim.x >> 5) + wid;
  if (node >= n) return;
  int head = lane >> 3;
  int c0   = lane * 8;
  float adv = ad[node * HEADS + head];
  int lo = rowptr[node], hi = rowptr[node + 1];

  // pass 1: running max of leaky-relu logits (self loop included)
  float e0 = as[node * HEADS + head] + adv;
  e0 = e0 > 0.f ? e0 : NEG_SLOPE * e0;
  float m = e0;
  for (int i = lo; i < hi; ++i) {
    int s = csr_src[i];
    float e = as[s * HEADS + head] + adv;
    e = e > 0.f ? e : NEG_SLOPE * e;
    m = fmaxf(m, e);
  }

  // pass 2: exp / sum / weighted accumulate
  float denom = 0.f;
  float acc[8];
#pragma unroll
  for (int j = 0; j < 8; ++j) acc[j] = 0.f;
  {
    float p = expf(e0 - m);
    denom += p;
    const float* hr = h + (size_t)node * HC + c0;
#pragma unroll
    for (int j = 0; j < 8; ++j) acc[j] += p * hr[j];
  }
  for (int i = lo; i < hi; ++i) {
    int s = csr_src[i];
    float e = as[s * HEADS + head] + adv;
    e = e > 0.f ? e : NEG_SLOPE * e;
    float p = expf(e - m);
    denom += p;
    const float* hr = h + (size_t)s * HC + c0;
#pragma unroll
    for (int j = 0; j < 8; ++j) acc[j] += p * hr[j];
  }
  float inv = 1.0f / (denom + 1e-16f);
  float* o = out + (size_t)node * HC + c0;
#pragma unroll
  for (int j = 0; j < 8; ++j) o[j] = acc[j] * inv + bias[c0 + j];
}

// ---------------- deterministic segment mean pool (batch is sorted) ----------------
__global__ void k_pool(const float* __restrict__ feat, const long long* __restrict__ batch,
                       float* __restrict__ pooled, int n) {
  int g = blockIdx.x;          // 32 graphs
  int ch = threadIdx.x;        // 256 channels
  // lower_bound(g) / lower_bound(g+1) — every thread computes (15 iters)
  int lo = 0, hi = n;
  while (lo < hi) { int mid = (lo + hi) >> 1; if (batch[mid] < (long long)g) lo = mid + 1; else hi = mid; }
  int b0 = lo;
  lo = 0; hi = n;
  while (lo < hi) { int mid = (lo + hi) >> 1; if (batch[mid] < (long long)(g + 1)) lo = mid + 1; else hi = mid; }
  int b1 = lo;
  float s = 0.f;
  for (int i = b0; i < b1; ++i) s += feat[(size_t)i * HC + ch];
  float cnt = (float)(b1 - b0);
  pooled[g * HC + ch] = s / fmaxf(cnt, 1.0f);
}

// ---------------- MLP tail: relu(P@lw1+lb1) -> relu(@lw2+lb2) ----------------
__global__ void k_tail(const float* __restrict__ pooled, const float* __restrict__ lw1,
                       const float* __restrict__ lb1, const float* __restrict__ lw2,
                       const float* __restrict__ lb2, float* __restrict__ out) {
  __shared__ float t1[32 * 128];
  int tid = threadIdx.x;
  for (int i = tid; i < 32 * 128; i += blockDim.x) {
    int g = i >> 7, j = i & 127;
    float s = lb1[j];
    const float* pr = pooled + g * HC;
#pragma unroll 8
    for (int k = 0; k < HC; ++k) s += pr[k] * lw1[k * 128 + j];
    t1[i] = s > 0.f ? s : 0.f;
  }
  __syncthreads();
  for (int i = tid; i < 32 * 10; i += blockDim.x) {
    int g = i / 10, j = i % 10;
    float s = lb2[j];
    const float* tr = t1 + g * 128;
#pragma unroll 8
    for (int k = 0; k < 128; ++k) s += tr[k] * lw2[k * 10 + j];
    out[i] = s > 0.f ? s : 0.f;
  }
}

// ---------------------------------------------------------------------------

static inline size_t align256(size_t x) { return (x + 255) & ~(size_t)255; }

extern "C" void kernel_launch(void* const* d_in, const int* in_sizes, int n_in,
                              void* d_out, int out_size, void* d_ws, size_t ws_size,
                              hipStream_t stream) {
  const float*     x      = (const float*)d_in[0];       // [N,62]
  const float*     pos    = (const float*)d_in[1];       // [N,2]
  const long long* eidx   = (const long long*)d_in[2];   // [2,E] int64
  const long long* batch  = (const long long*)d_in[3];   // [N]   int64 (sorted)
  const float*     W1     = (const float*)d_in[4];
  const float*     a_src1 = (const float*)d_in[5];
  const float*     a_dst1 = (const float*)d_in[6];
  const float*     b1     = (const float*)d_in[7];
  const float*     W2     = (const float*)d_in[8];
  const float*     a_src2 = (const float*)d_in[9];
  const float*     a_dst2 = (const float*)d_in[10];
  const float*     b2     = (const float*)d_in[11];
  const float*     lw1    = (const float*)d_in[12];
  const float*     lb1    = (const float*)d_in[13];
  const float*     lw2    = (const float*)d_in[14];
  const float*     lb2    = (const float*)d_in[15];
  float* out = (float*)d_out;

  const int N = in_sizes[0] / 62;     // 20000
  const int E = in_sizes[2] / 2;      // 320000
  const long long* srcs = eidx;       // row 0
  const long long* dsts = eidx + E;   // row 1

  // workspace carve-up
  char* ws = (char*)d_ws;
  size_t off = 0;
  float* X0     = (float*)(ws + off); off += align256((size_t)N * 64 * 4);
  float* hbuf   = (float*)(ws + off); off += align256((size_t)N * HC * 4);
  float* obuf   = (float*)(ws + off); off += align256((size_t)N * HC * 4);
  float* as     = (float*)(ws + off); off += align256((size_t)N * HEADS * 4);
  float* ad     = (float*)(ws + off); off += align256((size_t)N * HEADS * 4);
  int*   cnt    = (int*)  (ws + off); off += align256((size_t)N * 4);
  int*   rowptr = (int*)  (ws + off); off += align256((size_t)(N + 1) * 4);
  int*   cursor = (int*)  (ws + off); off += align256((size_t)N * 4);
  int*   csr    = (int*)  (ws + off); off += align256((size_t)E * 4);
  float* pooled = (float*)(ws + off); off += align256((size_t)32 * HC * 4);
  (void)ws_size; (void)n_in; (void)out_size;

  const int T = 256;
  // 0) concat inputs
  k_concat<<<(N * 64 + T - 1) / T, T, 0, stream>>>(x, pos, X0, N);
  // 1) CSR build over destinations
  k_zero_i32<<<(N + T - 1) / T, T, 0, stream>>>(cnt, N);
  k_count<<<(E + T - 1) / T, T, 0, stream>>>(dsts, cnt, E);
  k_scan<<<1, 256, 0, stream>>>(cnt, rowptr, N);
  k_copy_i32<<<(N + T - 1) / T, T, 0, stream>>>(rowptr, cursor, N);
  k_fill<<<(E + T - 1) / T, T, 0, stream>>>(srcs, dsts, cursor, csr, E);

  // GEMM grid: one 16-row strip per block (4 waves x 64 cols)
  const int gemm_blocks = (N + 15) / 16;

  // 2) GAT layer 1
  k_gemm_wmma<<<gemm_blocks, 128, 0, stream>>>(X0, W1, hbuf, N, 64);
  k_alpha<<<(N * HEADS + T - 1) / T, T, 0, stream>>>(hbuf, a_src1, a_dst1, as, ad, N);
  k_aggregate<<<(N + 3) / 4, 128, 0, stream>>>(hbuf, as, ad, rowptr, csr, b1, obuf, N);

  // 3) GAT layer 2 (input obuf, output back into obuf after hbuf consumed)
  k_gemm_wmma<<<gemm_blocks, 128, 0, stream>>>(obuf, W2, hbuf, N, HC);
  k_alpha<<<(N * HEADS + T - 1) / T, T, 0, stream>>>(hbuf, a_src2, a_dst2, as, ad, N);
  k_aggregate<<<(N + 3) / 4, 128, 0, stream>>>(hbuf, as, ad, rowptr, csr, b2, obuf, N);

  // 4) per-graph mean pool (deterministic, batch sorted)
  k_pool<<<32, 256, 0, stream>>>(obuf, batch, pooled, N);

  // 5) MLP tail
  k_tail<<<1, 256, 0, stream>>>(pooled, lw1, lb1, lw2, lb2, out);
}